// ResidualGNNBlock_54176717472256
// MI455X (gfx1250) — compile-verified
//
#include <hip/hip_runtime.h>
#include <hip/hip_bf16.h>

// ---------------- problem constants (match reference) ----------------
static constexpr int NN = 100000;   // nodes
static constexpr int EE = 500000;   // edges
static constexpr int CC = 128;      // channels
static constexpr float LN_EPS = 1e-5f;

typedef __attribute__((ext_vector_type(2))) float v2f;
typedef __attribute__((ext_vector_type(8))) float v8f;

// ---------------- gcn_norm kernels ----------------
__global__ void deg_init_kernel(float* __restrict__ deg, int n) {
    int i = blockIdx.x * blockDim.x + threadIdx.x;
    if (i < n) deg[i] = 1.0f;  // self loop contributes 1
}

__global__ void deg_count_kernel(const int* __restrict__ dst, float* __restrict__ deg, int e) {
    int i = blockIdx.x * blockDim.x + threadIdx.x;
    if (i < e) atomicAdd(&deg[dst[i]], 1.0f);
}

__global__ void deg_finalize_kernel(float* __restrict__ deg, int n) {
    int i = blockIdx.x * blockDim.x + threadIdx.x;
    if (i < n) deg[i] = rsqrtf(deg[i]);  // deg >= 1 always (self loops)
}

__global__ void edge_norm_kernel(const int* __restrict__ src, const int* __restrict__ dst,
                                 const float* __restrict__ ew, const float* __restrict__ dis,
                                 float* __restrict__ wn, int e) {
    int i = blockIdx.x * blockDim.x + threadIdx.x;
    if (i < e) wn[i] = dis[src[i]] * ew[i] * dis[dst[i]];
}

// ---------------- WMMA GEMM with fused aggregation-init ----------------
// H[N,128] = X[N,128] @ W[128,128]          (kept for the edge gather)
// AggInit[i,:] = bias[:] + dis[i]^2 * H[i,:] (self-loop + bias, atomics land on top)
//
// One block = 256 threads = 8 waves. Block b covers rows [16b,16b+16);
// wave w covers columns [16w,16w+16). K loop in steps of 4 via
// V_WMMA_F32_16X16X4_F32 (fp32 in/out, matches reference numerics).
// Weight matrix staged into LDS with GLOBAL_LOAD_ASYNC_TO_LDS_B128 (ASYNCcnt).
__global__ __launch_bounds__(256) void gemm_wmma_kernel(const float* __restrict__ X,
                                                        const float* __restrict__ W,
                                                        float* __restrict__ H,
                                                        const float* __restrict__ bias,
                                                        const float* __restrict__ dis,
                                                        float* __restrict__ AggInit) {
    __shared__ float sW[CC * CC];  // 64 KB of the 320 KB/WGP LDS

    const int tid = threadIdx.x;

    // --- async stage of the full 128x128 weight tile into LDS ---
    // 256 threads x 16 iterations x 16B = 64 KB, bypassing VGPRs entirely.
    {
        unsigned lbase = (unsigned)(size_t)(&sW[0]);  // low 32 bits = LDS offset
        unsigned long long gbase = (unsigned long long)(size_t)W;
        #pragma unroll
        for (int i = 0; i < 16; ++i) {
            unsigned off = (unsigned)(tid + i * 256) * 16u;
            unsigned la = lbase + off;
            unsigned long long ga = gbase + off;
            asm volatile("global_load_async_to_lds_b128 %0, %1, off"
                         :: "v"(la), "v"(ga) : "memory");
        }
        asm volatile("s_wait_asynccnt 0x0" ::: "memory");
    }
    __syncthreads();

    const int wave  = tid >> 5;          // 0..7 -> column tile
    const int lane  = tid & 31;
    const int m0    = blockIdx.x * 16;   // N = 100000 is divisible by 16
    const int n0    = wave * 16;
    const int half  = lane >> 4;         // 0: lanes 0-15, 1: lanes 16-31
    const int khalf = half * 2;          // A/B VGPR0 holds K=khalf, VGPR1 K=khalf+1
    const int mrow  = m0 + (lane & 15);
    const int ncol  = n0 + (lane & 15);

    const float* xrow = X + (size_t)mrow * CC;
    // hint the streamed A rows into cache (global_prefetch_b8)
    __builtin_prefetch(xrow, 0, 1);

    v8f acc = {};
    #pragma unroll 8
    for (int k = 0; k < CC; k += 4) {
        v2f a, b;
        a.x = xrow[k + khalf];
        a.y = xrow[k + khalf + 1];
        b.x = sW[(k + khalf) * CC + ncol];
        b.y = sW[(k + khalf + 1) * CC + ncol];
        // 8 args: (neg_a, A, neg_b, B, c_mod, C, reuse_a, reuse_b)
        acc = __builtin_amdgcn_wmma_f32_16x16x4_f32(
            false, a, false, b, (short)0, acc, false, false);
    }

    // C/D layout: VGPR r -> row m0 + r + 8*half, col = ncol
    const int rbase = m0 + 8 * half;
    const float bcol = bias[ncol];
    float* hp = H + (size_t)rbase * CC + ncol;
    float* ap = AggInit + (size_t)rbase * CC + ncol;
    #pragma unroll
    for (int r = 0; r < 8; ++r) {
        float dv = dis[rbase + r];
        float hv = acc[r];
        hp[(size_t)r * CC] = hv;
        ap[(size_t)r * CC] = bcol + dv * dv * hv;  // self-loop weight = dis^2
    }
}

// ---------------- edge scatter ----------------
// Out[dst,:] += w * H[src,:]  -- one wave per edge, 4 channels/lane, coalesced atomics
__global__ __launch_bounds__(256) void agg_edges_kernel(const int* __restrict__ src,
                                                        const int* __restrict__ dst,
                                                        const float* __restrict__ wn,
                                                        const float* __restrict__ H,
                                                        float* __restrict__ Out, int e) {
    int edge = blockIdx.x * 8 + (threadIdx.x >> 5);
    if (edge >= e) return;
    int lane = threadIdx.x & 31;
    int s = src[edge];
    int d = dst[edge];
    float w = wn[edge];
    const float* hs = H + (size_t)s * CC;
    float* od = Out + (size_t)d * CC;
    #pragma unroll
    for (int j = 0; j < 4; ++j) {
        int c = lane + 32 * j;
        atomicAdd(&od[c], w * hs[c]);
    }
}

// ---------------- LayerNorm + exact GELU (+ optional residual) ----------------
// One wave (32 lanes) per row of 128; wave32 shuffle reductions.
__global__ __launch_bounds__(256) void ln_gelu_kernel(const float* __restrict__ In,
                                                      const float* __restrict__ gamma,
                                                      const float* __restrict__ beta,
                                                      const float* __restrict__ residual,
                                                      float* __restrict__ Out, int n) {
    int row = blockIdx.x * 8 + (threadIdx.x >> 5);
    if (row >= n) return;
    int lane = threadIdx.x & 31;
    const float* p = In + (size_t)row * CC;

    float v[4];
    float s = 0.f;
    #pragma unroll
    for (int j = 0; j < 4; ++j) {
        v[j] = p[lane + 32 * j];
        s += v[j];
    }
    #pragma unroll
    for (int o = 16; o > 0; o >>= 1) s += __shfl_xor(s, o, 32);
    float mu = s * (1.0f / CC);

    float vs = 0.f;
    #pragma unroll
    for (int j = 0; j < 4; ++j) {
        float d = v[j] - mu;
        vs += d * d;
    }
    #pragma unroll
    for (int o = 16; o > 0; o >>= 1) vs += __shfl_xor(vs, o, 32);
    float inv = rsqrtf(vs * (1.0f / CC) + LN_EPS);

    float* op = Out + (size_t)row * CC;
    const float* rp = residual ? residual + (size_t)row * CC : nullptr;
    #pragma unroll
    for (int j = 0; j < 4; ++j) {
        int c = lane + 32 * j;
        float t = (v[j] - mu) * inv * gamma[c] + beta[c];
        float g = 0.5f * t * (1.0f + erff(t * 0.70710678118654752f));  // exact gelu
        op[c] = rp ? (g + rp[c]) : g;
    }
}

// ---------------- launcher ----------------
extern "C" void kernel_launch(void* const* d_in, const int* in_sizes, int n_in,
                              void* d_out, int out_size, void* d_ws, size_t ws_size,
                              hipStream_t stream) {
    const float* x   = (const float*)d_in[0];
    const int*   ei  = (const int*)d_in[1];
    const float* ew  = (const float*)d_in[2];
    const float* W1  = (const float*)d_in[3];
    const float* b1  = (const float*)d_in[4];
    const float* g1  = (const float*)d_in[5];
    const float* be1 = (const float*)d_in[6];
    const float* W2  = (const float*)d_in[7];
    const float* b2  = (const float*)d_in[8];
    const float* g2  = (const float*)d_in[9];
    const float* be2 = (const float*)d_in[10];
    float* out = (float*)d_out;

    const int* src = ei;        // edge_index row 0
    const int* dst = ei + EE;   // edge_index row 1

    // workspace layout (floats): dis[N] | wn[E] | bufA[N*C] | bufB[N*C]
    float* dis  = (float*)d_ws;
    float* wn   = dis + NN;
    float* bufA = wn + EE;
    float* bufB = bufA + (size_t)NN * CC;

    const int T = 256;

    // --- gcn_norm ---
    deg_init_kernel<<<(NN + T - 1) / T, T, 0, stream>>>(dis, NN);
    deg_count_kernel<<<(EE + T - 1) / T, T, 0, stream>>>(dst, dis, EE);
    deg_finalize_kernel<<<(NN + T - 1) / T, T, 0, stream>>>(dis, NN);
    edge_norm_kernel<<<(EE + T - 1) / T, T, 0, stream>>>(src, dst, ew, dis, wn, EE);

    // --- layer 1: GEMM (+ fused bias/self-loop init into bufB), scatter, LN+GELU ---
    gemm_wmma_kernel<<<NN / 16, T, 0, stream>>>(x, W1, bufA, b1, dis, bufB);
    agg_edges_kernel<<<(EE + 7) / 8, T, 0, stream>>>(src, dst, wn, bufA, bufB, EE);
    ln_gelu_kernel<<<(NN + 7) / 8, T, 0, stream>>>(bufB, g1, be1, nullptr, bufB, NN);

    // --- layer 2 ---
    gemm_wmma_kernel<<<NN / 16, T, 0, stream>>>(bufB, W2, bufA, b2, dis, bufB);
    agg_edges_kernel<<<(EE + 7) / 8, T, 0, stream>>>(src, dst, wn, bufA, bufB, EE);

    // --- final LN + GELU + residual ---
    ln_gelu_kernel<<<(NN + 7) / 8, T, 0, stream>>>(bufB, g2, be2, x, out, NN);
}